// FP_Solver_3685081940115
// MI455X (gfx1250) — compile-verified
//
#include <hip/hip_runtime.h>
#include <hip/hip_bf16.h>

typedef __attribute__((ext_vector_type(2))) float v2f;
typedef __attribute__((ext_vector_type(8))) float v8f;

#define WMMA_F32_16x16x4(a, b, c) \
    __builtin_amdgcn_wmma_f32_16x16x4_f32(false, (a), false, (b), (short)0, (c), false, false)

// im2col offset for conv1 (clamped: k>=45 maps to last valid tap; A is
// zero-padded there so the product contributes nothing).
__device__ __forceinline__ int koff1(int k) {
    int kk = (k < 45) ? k : 44;
    int ci = kk / 9, r = kk - ci * 9;
    int dy = r / 3,  dx = r - dy * 3;
    return ci * 400 + dy * 20 + dx;
}

// LDS partition (in floats)
#define OFF_XT   0        // 5*20*20              = 2000
#define OFF_W1   2000     // 64*48 (k zero-pad)   = 3072
#define OFF_W2   5072     // 6*576 (tap-permuted, row 5 = zeros) = 3456
#define OFF_B1   8528     // 64
#define OFF_B2   8592     // 8
#define OFF_HT   8600     // 64*18*18             = 20736
#define SMEM_FLOATS 29336

// ---------------------------------------------------------------------------
// Fused phi_r: conv(5->64,3x3)+b1+relu -> conv(64->5,3x3)+b2,
// optional fixed-point combine out = o*(1-mask) + yobs*mask.
// ---------------------------------------------------------------------------
__global__ __launch_bounds__(256)
void phi_fused_kernel(const float* __restrict__ xin, float* __restrict__ out,
                      const float* __restrict__ W1g, const float* __restrict__ b1g,
                      const float* __restrict__ W2g, const float* __restrict__ b2g,
                      const int* __restrict__ mask, const float* __restrict__ yobs,
                      int apply_mask)
{
    extern __shared__ float smem[];
    float* xt  = smem + OFF_XT;
    float* w1p = smem + OFF_W1;
    float* w2p = smem + OFF_W2;
    float* bs1 = smem + OFF_B1;
    float* bs2 = smem + OFF_B2;
    float* ht  = smem + OFF_HT;

    const int tid  = threadIdx.x;
    const int lane = tid & 31;
    const int wv   = tid >> 5;
    const int half = lane >> 4;    // WMMA lane-half (adds +2 to k, +8 to D row)
    const int lm   = lane & 15;

    const int OX = blockIdx.x * 16;   // NY (j) origin
    const int OY = blockIdx.y * 16;   // NX (i) origin
    const int b  = blockIdx.z;

    // ---- setup: weights (padded / permuted), biases, x tile (halo 2) ----
    for (int p = tid; p < 3072; p += 256) {            // W1 -> [64][48], zero pad
        int ch = p / 48, k = p - ch * 48;
        w1p[p] = (k < 45) ? W1g[ch * 45 + k] : 0.f;
    }
    for (int p = tid; p < 3456; p += 256) {            // W2 -> [6][tap*64+ci]
        int row = p / 576, q = p - row * 576;
        int tap = q >> 6, ci = q & 63;
        w2p[p] = (row < 5) ? W2g[row * 576 + ci * 9 + tap] : 0.f;
    }
    if (tid < 64) bs1[tid] = b1g[tid];
    if (tid < 5)  bs2[tid] = b2g[tid];
    for (int p = tid; p < 2000; p += 256) {
        int c  = p / 400; int rr = p - c * 400;
        int yy = rr / 20; int xx = rr - yy * 20;
        int gi = OY + yy - 2, gj = OX + xx - 2;
        float v = 0.f;
        if (gi >= 0 && gi < 512 && gj >= 0 && gj < 512)
            v = xin[(((size_t)b * 5 + c) * 512 + gi) * 512 + gj];
        xt[p] = v;
    }
    __syncthreads();

    // ---- Stage 1: H[64,324] = relu(W1[64,45] @ im2col(x)[45,324] + b1) ----
    // Per n-tile: one B fragment feeds 4 M-tile accumulators.
    // A fragments loaded as aligned v2f -> ds_load_b64 with immediate offsets
    // from a single per-lane base pointer.
    const float* wa1 = w1p + (lm * 48 + 2 * half);     // 8B-aligned base
    for (int nt = wv; nt < 21; nt += 8) {
        const int n = nt * 16 + lm;                    // pixel in 18x18 h-tile
        const bool nvalid = (n < 324);
        const int hy = n / 18, hx = n - hy * 18;
        const int base1 = nvalid ? (hy * 20 + hx) : 0; // clamp: loads stay in-range
        const float* xb = xt + base1;
        v8f a0 = {}, a1 = {}, a2 = {}, a3 = {};
        #pragma unroll
        for (int k0 = 0; k0 < 45; k0 += 4) {
            v2f bb;
            bb.x = xb[half ? koff1(k0 + 2) : koff1(k0 + 0)];
            bb.y = xb[half ? koff1(k0 + 3) : koff1(k0 + 1)];
            v2f w0 = *(const v2f*)(wa1 + 0 * 768 + k0);
            v2f w1 = *(const v2f*)(wa1 + 1 * 768 + k0);
            v2f w2 = *(const v2f*)(wa1 + 2 * 768 + k0);
            v2f w3 = *(const v2f*)(wa1 + 3 * 768 + k0);
            a0 = WMMA_F32_16x16x4(w0, bb, a0);
            a1 = WMMA_F32_16x16x4(w1, bb, a1);
            a2 = WMMA_F32_16x16x4(w2, bb, a2);
            a3 = WMMA_F32_16x16x4(w3, bb, a3);
        }
        if (nvalid) {
            #pragma unroll
            for (int r = 0; r < 8; ++r) {
                const int rr = r + 8 * half;           // D row within 16
                float v0 = a0[r] + bs1[ 0 + rr];
                float v1 = a1[r] + bs1[16 + rr];
                float v2 = a2[r] + bs1[32 + rr];
                float v3 = a3[r] + bs1[48 + rr];
                ht[( 0 + rr) * 324 + n] = v0 > 0.f ? v0 : 0.f;
                ht[(16 + rr) * 324 + n] = v1 > 0.f ? v1 : 0.f;
                ht[(32 + rr) * 324 + n] = v2 > 0.f ? v2 : 0.f;
                ht[(48 + rr) * 324 + n] = v3 > 0.f ? v3 : 0.f;
            }
        }
    }
    __syncthreads();

    // ---- Stage 2: O[5(pad16),256] = W2 @ im2col(H), K reordered tap*64+ci ----
    // Each wave: output rows oy0, oy0+1; A fragment shared by both accumulators.
    const int rowa = (lm < 5) ? lm : 5;                // row 5 of w2p is zeros
    const float* wa2 = w2p + (rowa * 576 + 2 * half);  // 8B-aligned base
    const int oy0  = wv * 2;
    const float* hbp = ht + (oy0 * 18 + lm + half * 648);   // + half*2*324
    v8f cA = {}, cB = {};
    #pragma unroll
    for (int tap = 0; tap < 9; ++tap) {
        const int tdy = tap / 3, tdx = tap - tdy * 3;
        const int toff = tdy * 18 + tdx;
        #pragma unroll
        for (int ci0 = 0; ci0 < 64; ci0 += 4) {
            v2f wa = *(const v2f*)(wa2 + tap * 64 + ci0);
            const int bo = toff + ci0 * 324;
            v2f bA, bB;
            bA.x = hbp[bo];           bA.y = hbp[bo + 324];
            bB.x = hbp[bo + 18];      bB.y = hbp[bo + 324 + 18];
            cA = WMMA_F32_16x16x4(wa, bA, cA);
            cB = WMMA_F32_16x16x4(wa, bB, cB);
        }
    }
    // D rows 0..4 on lanes 0..15 are the real T channels
    if (half == 0) {
        const int gj = OX + lm;
        #pragma unroll
        for (int r = 0; r < 5; ++r) {
            const size_t plane = (((size_t)b * 5 + r) * 512);
            float vA = cA[r] + bs2[r];
            float vB = cB[r] + bs2[r];
            size_t iA = (plane + (OY + oy0    )) * 512 + gj;
            size_t iB = (plane + (OY + oy0 + 1)) * 512 + gj;
            if (apply_mask) {
                float mA = (float)mask[iA], mB = (float)mask[iB];
                vA = vA * (1.f - mA) + yobs[iA] * mA;
                vB = vB * (1.f - mB) + yobs[iB] * mB;
            }
            out[iA] = vA;
            out[iB] = vB;
        }
    }
}

// ---------------------------------------------------------------------------
// Cost row partials: per-batch sums of (gt-x)^2, mask*(x-yobs)^2,
// and x^T Q x via the 7-point stencil (6.01*x^2 - 2*x*forward-neighbors).
// ---------------------------------------------------------------------------
__global__ __launch_bounds__(256)
void cost_partial_kernel(const float* __restrict__ xk, const float* __restrict__ gt,
                         const float* __restrict__ yobs, const int* __restrict__ mask,
                         float* __restrict__ partials /* [B][gridDim.x][3] */)
{
    const int b = blockIdx.y;
    const size_t fsz = (size_t)5 * 512 * 512;
    const float* x  = xk   + (size_t)b * fsz;
    const float* g  = gt   + (size_t)b * fsz;
    const float* yo = yobs + (size_t)b * fsz;
    const int*   mk = mask + (size_t)b * fsz;

    float smse = 0.f, sdy = 0.f, sq = 0.f;
    for (size_t i = (size_t)blockIdx.x * blockDim.x + threadIdx.x; i < fsz;
         i += (size_t)gridDim.x * blockDim.x) {
        float v = x[i];
        float d = g[i] - v;       smse += d * d;
        float mf = (float)mk[i];
        float dy = v - yo[i];     sdy  += mf * dy * dy;
        float q = 6.01f * v * v;
        size_t t  = i >> 18;                 // / (512*512)
        size_t rr = i & ((size_t)(1 << 18) - 1);
        size_t ii = rr >> 9;
        size_t jj = rr & 511;
        if (t  < 4)   q -= 2.f * v * x[i + (size_t)512 * 512];
        if (ii < 511) q -= 2.f * v * x[i + 512];
        if (jj < 511) q -= 2.f * v * x[i + 1];
        sq += q;
    }

    __shared__ float red[3 * 256];
    red[threadIdx.x]       = smse;
    red[256 + threadIdx.x] = sdy;
    red[512 + threadIdx.x] = sq;
    __syncthreads();
    for (int s = 128; s > 0; s >>= 1) {
        if (threadIdx.x < s) {
            red[threadIdx.x]       += red[threadIdx.x + s];
            red[256 + threadIdx.x] += red[256 + threadIdx.x + s];
            red[512 + threadIdx.x] += red[512 + threadIdx.x + s];
        }
        __syncthreads();
    }
    if (threadIdx.x == 0) {
        float* p = partials + ((size_t)b * gridDim.x + blockIdx.x) * 3;
        p[0] = red[0]; p[1] = red[256]; p[2] = red[512];
    }
}

__global__ __launch_bounds__(256)
void cost_final_kernel(const float* __restrict__ partials, int nblk,
                       float* __restrict__ cmp, int row)
{
    const int b = blockIdx.x;
    float smse = 0.f, sdy = 0.f, sq = 0.f;
    for (int i = threadIdx.x; i < nblk; i += 256) {
        const float* p = partials + ((size_t)b * nblk + i) * 3;
        smse += p[0]; sdy += p[1]; sq += p[2];
    }
    __shared__ float red[3 * 256];
    red[threadIdx.x]       = smse;
    red[256 + threadIdx.x] = sdy;
    red[512 + threadIdx.x] = sq;
    __syncthreads();
    for (int s = 128; s > 0; s >>= 1) {
        if (threadIdx.x < s) {
            red[threadIdx.x]       += red[threadIdx.x + s];
            red[256 + threadIdx.x] += red[256 + threadIdx.x + s];
            red[512 + threadIdx.x] += red[512 + threadIdx.x + s];
        }
        __syncthreads();
    }
    if (threadIdx.x == 0) {
        cmp[b * 10 + row * 2 + 0] = red[0] / 1310720.f;             // mean over T*NX*NY
        cmp[b * 10 + row * 2 + 1] = 1000.f * red[256] + red[512];   // dy_term + xQx
    }
}

// ---------------------------------------------------------------------------
extern "C" void kernel_launch(void* const* d_in, const int* in_sizes, int n_in,
                              void* d_out, int out_size, void* d_ws, size_t ws_size,
                              hipStream_t stream) {
    const float* gt   = (const float*)d_in[0];
    const float* x0   = (const float*)d_in[1];
    const float* yobs = (const float*)d_in[2];
    const int*   mask = (const int*)  d_in[3];
    const float* W1   = (const float*)d_in[4];
    const float* b1   = (const float*)d_in[5];
    const float* W2   = (const float*)d_in[6];
    const float* b2   = (const float*)d_in[7];
    // d_in[8..10] (Q COO) unneeded: Q is applied analytically as a stencil.
    // n_fp (d_in[11]) is the compile-time constant 4 in the reference.

    const size_t FIELD = (size_t)4 * 5 * 512 * 512;   // 5,242,880
    float* outx = (float*)d_out;
    float* cmp  = outx + FIELD;                        // 40 floats
    float* bufA = (float*)d_ws;                        // one ping-pong field
    float* part = bufA + FIELD;                        // [4][512][3] partials

    const dim3 convGrid(32, 32, 4);
    const int  convLds = SMEM_FLOATS * sizeof(float);  // ~117 KB dynamic LDS
    const int  NBLK = 512;
    const dim3 costGrid(NBLK, 4);

    auto cost = [&](const float* xk, int row) {
        cost_partial_kernel<<<costGrid, 256, 0, stream>>>(xk, gt, yobs, mask, part);
        cost_final_kernel<<<dim3(4), 256, 0, stream>>>(part, NBLK, cmp, row);
    };
    auto conv = [&](const float* in, float* out, int masked) {
        phi_fused_kernel<<<convGrid, 256, convLds, stream>>>(
            in, out, W1, b1, W2, b2, mask, yobs, masked);
    };

    // x -> out -> A -> out -> A -> out(final, unmasked); cost before each update
    cost(x0, 0);
    conv(x0,   outx, 1);
    cost(outx, 1);
    conv(outx, bufA, 1);
    cost(bufA, 2);
    conv(bufA, outx, 1);
    cost(outx, 3);
    conv(outx, bufA, 1);
    conv(bufA, outx, 0);   // final phi_r, no mask combine
    cost(outx, 4);
}